// OHEMMixupBCELoss_40372692582426
// MI455X (gfx1250) — compile-verified
//
#include <hip/hip_runtime.h>
#include <math.h>

#ifndef M_PI
#define M_PI 3.14159265358979323846
#endif

#define HIST1_BINS 2048   // bits [30:20]  (sign always 0 since loss >= 0)
#define HIST2_BINS 2048   // bits [19:9]
#define HIST3_BINS 512    // bits [8:0]

typedef __attribute__((ext_vector_type(2))) float v2f;
typedef __attribute__((ext_vector_type(8))) float v8f;

// loss = lam*bce(x,y1) + (1-lam)*bce(x,y2)
//      = relu(x) + log1p(exp(-|x|)) - x*(lam*y1 + (1-lam)*y2)
// clamped at 0 so the uint bit pattern is monotone in value.
__device__ __forceinline__ float mixed_bce(float x, float y1, float y2, float lam) {
  float ymix = __fmaf_rn(lam, y1 - y2, y2);
  float e = __expf(-fabsf(x));          // v_exp_f32
  float soft = __logf(1.0f + e);        // v_log_f32
  float l = fmaxf(x, 0.0f) - x * ymix + soft;
  return fmaxf(l, 0.0f);
}

__global__ void ohem_zero(unsigned* __restrict__ p, int n) {
  int i = blockIdx.x * blockDim.x + threadIdx.x;
  if (i < n) p[i] = 0u;
}

// Pass 1: compute loss, optionally store it, histogram bits[30:20].
__global__ void ohem_pass1(const float* __restrict__ yp, const float* __restrict__ t1,
                           const float* __restrict__ t2, const float* __restrict__ lamp,
                           float* __restrict__ lossbuf, int use_store,
                           unsigned* __restrict__ hist1, int n4) {
  __shared__ unsigned h[HIST1_BINS];
  for (int i = threadIdx.x; i < HIST1_BINS; i += blockDim.x) h[i] = 0u;
  __syncthreads();
  const float lam = lamp[0];
  const int stride = gridDim.x * blockDim.x;
  for (int i = blockIdx.x * blockDim.x + threadIdx.x; i < n4; i += stride) {
    float4 x = reinterpret_cast<const float4*>(yp)[i];
    float4 a = reinterpret_cast<const float4*>(t1)[i];
    float4 b = reinterpret_cast<const float4*>(t2)[i];
    float4 L;
    L.x = mixed_bce(x.x, a.x, b.x, lam);
    L.y = mixed_bce(x.y, a.y, b.y, lam);
    L.z = mixed_bce(x.z, a.z, b.z, lam);
    L.w = mixed_bce(x.w, a.w, b.w, lam);
    if (use_store) reinterpret_cast<float4*>(lossbuf)[i] = L;
    atomicAdd(&h[__float_as_uint(L.x) >> 20], 1u);
    atomicAdd(&h[__float_as_uint(L.y) >> 20], 1u);
    atomicAdd(&h[__float_as_uint(L.z) >> 20], 1u);
    atomicAdd(&h[__float_as_uint(L.w) >> 20], 1u);
  }
  __syncthreads();
  for (int i = threadIdx.x; i < HIST1_BINS; i += blockDim.x)
    if (h[i]) atomicAdd(&hist1[i], h[i]);
}

// Find the bin where the from-the-top cumulative count crosses k.
// sel layout: [0]=b1 [1]=cntGt1 [2]=b2 [3]=cntGt2 [4]=Tbits [5]=cntGtTotal, [8]=sum(f32)
__global__ void ohem_select(const unsigned* __restrict__ hist, unsigned* __restrict__ sel,
                            unsigned kTotal, int stage) {
  __shared__ unsigned chunk[256];
  __shared__ unsigned above[256];
  const int nbins = (stage == 1) ? HIST1_BINS : (stage == 2) ? HIST2_BINS : HIST3_BINS;
  const int per = nbins / 256;
  unsigned base = 0;
  if (stage == 2) base = sel[1];
  else if (stage == 3) base = sel[3];
  const unsigned kWant = kTotal - base;
  const int t = threadIdx.x;
  unsigned s = 0;
  for (int i = 0; i < per; ++i) s += hist[t * per + i];
  chunk[t] = s;
  __syncthreads();
  if (t == 0) {
    unsigned run = 0;
    for (int j = 255; j >= 0; --j) { above[j] = run; run += chunk[j]; }
  }
  __syncthreads();
  unsigned run = above[t];   // count of elements in bins strictly above this chunk
  for (int i = per - 1; i >= 0; --i) {
    const unsigned b = (unsigned)(t * per + i);
    const unsigned hv = hist[b];
    if (run < kWant && run + hv >= kWant) {   // exactly one thread hits this
      if (stage == 1)      { sel[0] = b; sel[1] = run; }
      else if (stage == 2) { sel[2] = b; sel[3] = base + run; }
      else {
        const unsigned T = (sel[0] << 20) | (sel[2] << 9) | b;  // exact 32-bit threshold
        sel[4] = T;
        sel[5] = base + run;                  // count strictly greater than T
      }
    }
    run += hv;
  }
}

// Passes 2/3: refine histogram inside the selected bin.
__global__ void ohem_pass23(const float* __restrict__ yp, const float* __restrict__ t1,
                            const float* __restrict__ t2, const float* __restrict__ lamp,
                            const float* __restrict__ lossbuf, int use_store,
                            const unsigned* __restrict__ sel, unsigned* __restrict__ hist,
                            int stage, int n4) {
  __shared__ unsigned h[HIST2_BINS];
  const int nbins = (stage == 2) ? HIST2_BINS : HIST3_BINS;
  for (int i = threadIdx.x; i < nbins; i += blockDim.x) h[i] = 0u;
  __syncthreads();
  unsigned matchVal, mshift, bshift, bmask;
  if (stage == 2) { matchVal = sel[0];                   mshift = 20; bshift = 9; bmask = HIST2_BINS - 1; }
  else            { matchVal = (sel[0] << 11) | sel[2];  mshift = 9;  bshift = 0; bmask = HIST3_BINS - 1; }
  const float lam = lamp[0];
  const int stride = gridDim.x * blockDim.x;
  for (int i = blockIdx.x * blockDim.x + threadIdx.x; i < n4; i += stride) {
    float4 L;
    if (use_store) {
      L = reinterpret_cast<const float4*>(lossbuf)[i];
    } else {
      float4 x = reinterpret_cast<const float4*>(yp)[i];
      float4 a = reinterpret_cast<const float4*>(t1)[i];
      float4 b = reinterpret_cast<const float4*>(t2)[i];
      L.x = mixed_bce(x.x, a.x, b.x, lam);
      L.y = mixed_bce(x.y, a.y, b.y, lam);
      L.z = mixed_bce(x.z, a.z, b.z, lam);
      L.w = mixed_bce(x.w, a.w, b.w, lam);
    }
    unsigned u;
    u = __float_as_uint(L.x); if ((u >> mshift) == matchVal) atomicAdd(&h[(u >> bshift) & bmask], 1u);
    u = __float_as_uint(L.y); if ((u >> mshift) == matchVal) atomicAdd(&h[(u >> bshift) & bmask], 1u);
    u = __float_as_uint(L.z); if ((u >> mshift) == matchVal) atomicAdd(&h[(u >> bshift) & bmask], 1u);
    u = __float_as_uint(L.w); if ((u >> mshift) == matchVal) atomicAdd(&h[(u >> bshift) & bmask], 1u);
  }
  __syncthreads();
  for (int i = threadIdx.x; i < nbins; i += blockDim.x)
    if (h[i]) atomicAdd(&hist[i], h[i]);
}

// Sum of all values strictly greater than T.  Wave reduction via
// V_WMMA_F32_16X16X4_F32: with B = all-ones, every column of D equals the
// sum of all entries of A, so sum(D regs in lane j) + same at lane j+16 = wave total.
__global__ void ohem_sum(const float* __restrict__ yp, const float* __restrict__ t1,
                         const float* __restrict__ t2, const float* __restrict__ lamp,
                         const float* __restrict__ lossbuf, int use_store,
                         const unsigned* __restrict__ sel, float* __restrict__ sumslot, int n4) {
  const unsigned T = sel[4];
  const float lam = lamp[0];
  float acc = 0.0f;
  const int stride = gridDim.x * blockDim.x;
  for (int i = blockIdx.x * blockDim.x + threadIdx.x; i < n4; i += stride) {
    float4 L;
    if (use_store) {
      L = reinterpret_cast<const float4*>(lossbuf)[i];
    } else {
      float4 x = reinterpret_cast<const float4*>(yp)[i];
      float4 a = reinterpret_cast<const float4*>(t1)[i];
      float4 b = reinterpret_cast<const float4*>(t2)[i];
      L.x = mixed_bce(x.x, a.x, b.x, lam);
      L.y = mixed_bce(x.y, a.y, b.y, lam);
      L.z = mixed_bce(x.z, a.z, b.z, lam);
      L.w = mixed_bce(x.w, a.w, b.w, lam);
    }
    acc += (__float_as_uint(L.x) > T) ? L.x : 0.0f;
    acc += (__float_as_uint(L.y) > T) ? L.y : 0.0f;
    acc += (__float_as_uint(L.z) > T) ? L.z : 0.0f;
    acc += (__float_as_uint(L.w) > T) ? L.w : 0.0f;
  }
  // ---- wave32 reduction with one WMMA (EXEC is all-1s here: uniform control flow) ----
  v2f a2; a2[0] = acc;  a2[1] = 0.0f;   // A: each lane contributes acc at one (m,k) slot
  v2f b2; b2[0] = 1.0f; b2[1] = 1.0f;   // B: all ones (layout-invariant)
  v8f c = {0.0f, 0.0f, 0.0f, 0.0f, 0.0f, 0.0f, 0.0f, 0.0f};
  c = __builtin_amdgcn_wmma_f32_16x16x4_f32(false, a2, false, b2, (short)0, c, false, false);
  float s = c[0] + c[1] + c[2] + c[3] + c[4] + c[5] + c[6] + c[7];
  s += __shfl_xor(s, 16, 32);           // combine the two row-halves -> full wave sum
  if ((threadIdx.x & 31u) == 0u) atomicAdd(sumslot, s);
}

__global__ void ohem_final(const unsigned* __restrict__ sel, const float* __restrict__ sumslot,
                           float* __restrict__ out, unsigned k) {
  if (threadIdx.x == 0 && blockIdx.x == 0) {
    const float sum_gt = *sumslot;
    const unsigned T = sel[4];
    const unsigned cgt = sel[5];          // count strictly greater than T (< k)
    const float thr = __uint_as_float(T); // exact threshold value; ties contribute thr each
    out[0] = (float)(((double)sum_gt + (double)(k - cgt) * (double)thr) / (double)k);
  }
}

extern "C" void kernel_launch(void* const* d_in, const int* in_sizes, int n_in,
                              void* d_out, int out_size, void* d_ws, size_t ws_size,
                              hipStream_t stream) {
  const float* yp   = (const float*)d_in[0];
  const float* t1   = (const float*)d_in[1];
  const float* t2   = (const float*)d_in[2];
  const float* lamp = (const float*)d_in[3];
  float* out = (float*)d_out;
  const int n  = in_sizes[0];
  const int n4 = n >> 2;                 // n = 2^24, divisible by 4

  // Mirror the Python: rate = 0.125 + 0.875/2*(cos(pi*0.5)+1); k = int(rate*n)
  const double rate = 0.125 + (1.0 - 0.125) / 2.0 * (cos(M_PI * 0.5) + 1.0);
  const unsigned k = (unsigned)(rate * (double)n);   // == 9437184 for n=2^24

  unsigned char* w = (unsigned char*)d_ws;
  unsigned* hist1 = (unsigned*)w;
  unsigned* hist2 = hist1 + HIST1_BINS;
  unsigned* hist3 = hist2 + HIST2_BINS;
  unsigned* sel   = hist3 + HIST3_BINS;
  float* sumslot  = (float*)(sel + 8);
  float* lossbuf  = (float*)(w + 32768);
  const int use_store = (ws_size >= (size_t)32768 + (size_t)n * sizeof(float)) ? 1 : 0;

  const int B = 256;       // 8 waves per workgroup on wave32
  const int G = 8192;      // grid-stride; ~2 float4 per thread

  const int nz = HIST1_BINS + HIST2_BINS + HIST3_BINS + 16;
  ohem_zero  <<<(nz + B - 1) / B, B, 0, stream>>>(hist1, nz);
  ohem_pass1 <<<G, B, 0, stream>>>(yp, t1, t2, lamp, lossbuf, use_store, hist1, n4);
  ohem_select<<<1, 256, 0, stream>>>(hist1, sel, k, 1);
  ohem_pass23<<<G, B, 0, stream>>>(yp, t1, t2, lamp, lossbuf, use_store, sel, hist2, 2, n4);
  ohem_select<<<1, 256, 0, stream>>>(hist2, sel, k, 2);
  ohem_pass23<<<G, B, 0, stream>>>(yp, t1, t2, lamp, lossbuf, use_store, sel, hist3, 3, n4);
  ohem_select<<<1, 256, 0, stream>>>(hist3, sel, k, 3);
  ohem_sum   <<<G, B, 0, stream>>>(yp, t1, t2, lamp, lossbuf, use_store, sel, sumslot, n4);
  ohem_final <<<1, 1, 0, stream>>>(sel, sumslot, out, k);
}